// PGM_LDS_Simple_60017872995068
// MI455X (gfx1250) — compile-verified
//
#include <hip/hip_runtime.h>
#include <hip/hip_bf16.h>

// ---------------------------------------------------------------------------
// PGM scalar-Kalman chain (T=24, D=262144): forward info filter + fused
// RTS smoother / backward sampler, per-dim KL reduced deterministically.
// CDNA5 path: async global->LDS staging (ASYNCcnt, non-temporal) of each
// block's working set, compute from LDS, wave32 blocks, NT output streams.
// VALU is the binding constraint (memory roof ~7 us): threefry pairing
// (12 evals -> 24 words) and all 24 erfinv conversions executed in the
// DMA shadow, OCML erfinvf (leanest codegen per round-2/3 histograms).
// ---------------------------------------------------------------------------

#define T_STEPS 24
#define S_ROWS  12
#define BLOCK   256

#define LOG_2PI_F 1.8378770664093453f
#define EPS_F     1e-6f
#define SQRT2_F   1.4142135623730951f

// LDS staging layout (float indices)
#define LDS_RJ  0
#define LDS_RH  (T_STEPS * BLOCK)
#define LDS_AO  (2 * T_STEPS * BLOCK)
#define LDS_LO  (2 * T_STEPS * BLOCK + S_ROWS * BLOCK)
#define LDS_BO  (2 * T_STEPS * BLOCK + 2 * S_ROWS * BLOCK)
#define LDS_TOT (2 * T_STEPS * BLOCK + 3 * S_ROWS * BLOCK)  // 21504 f32 = 84 KB

typedef __attribute__((address_space(3))) void lds_void_t;

__device__ __forceinline__ unsigned int lds_byte_addr(const void* p) {
  // generic -> LDS addrspace cast, then ptrtoint: byte offset in group segment
  return (unsigned int)(unsigned long long)(lds_void_t*)(p);
}

__device__ __forceinline__ void async_load_b128(unsigned int lds_addr,
                                                const void* gaddr) {
  // per-lane 16B DMA from the global address gaddr to the LDS address lds_addr; tracked by ASYNCcnt.
  // Data is single-touch -> non-temporal hint keeps L2 clean.
  asm volatile("global_load_async_to_lds_b128 %0, %1, off th:TH_LOAD_NT"
               :
               : "v"(lds_addr), "v"(gaddr)
               : "memory");
}

__device__ __forceinline__ void wait_asynccnt0() {
  asm volatile("s_wait_asynccnt 0x0" ::: "memory");
}

// One threefry2x32 evaluation (key = (0,0)), 20 rounds, yields BOTH lanes of
// the counter pair (x0, x1). JAX's flat layout pairs indices (i, i + N/2),
// i.e. times (t, t+12) of the same dimension -> 12 evals cover all 24 eps.
__device__ __forceinline__ void threefry2x32_00(unsigned int& x0,
                                                unsigned int& x1) {
  const unsigned int ks[3] = {0u, 0u, 0x1BD11BDAu};
  const unsigned int rot[2][4] = {{13u, 15u, 26u, 6u}, {17u, 29u, 16u, 24u}};
  // x0 += ks[0]; x1 += ks[1];  (both zero for key 0)
#pragma unroll
  for (int g = 0; g < 5; ++g) {
#pragma unroll
    for (int j = 0; j < 4; ++j) {
      const unsigned int r = rot[g & 1][j];
      x0 += x1;
      x1 = (x1 << r) | (x1 >> (32u - r));
      x1 ^= x0;
    }
    x0 += ks[(g + 1) % 3];
    x1 += ks[(g + 2) % 3] + (unsigned int)(g + 1);
  }
}

// bits -> uniform(-1+eps, 1) -> sqrt(2)*erfinv  (matches jax.random.normal)
__device__ __forceinline__ float bits_to_normal(unsigned int bits) {
  const float f = __uint_as_float((bits >> 9) | 0x3f800000u) - 1.0f;  // [0,1)
  const float lo = -0.99999994f;  // nextafter(-1, 0)
  float u = f * (1.0f - lo) + lo;
  u = fmaxf(lo, u);
  return SQRT2_F * erfinvf(u);
}

__global__ __launch_bounds__(BLOCK) void pgm_chain_kernel(
    const float* __restrict__ recog_J, const float* __restrict__ recog_h,
    const float* __restrict__ loc, const float* __restrict__ Tau,
    const float* __restrict__ Lambda, const float* __restrict__ A,
    const float* __restrict__ b, float* __restrict__ z_out,
    float* __restrict__ Ex_out, float* __restrict__ Exx_out,
    float* __restrict__ kl_partial, int D) {
  __shared__ float smem[LDS_TOT];
  const int tid = threadIdx.x;
  const int d0 = blockIdx.x * BLOCK;
  const int d = d0 + tid;
  const unsigned int base = lds_byte_addr(smem);

  // ---- async global -> LDS staging of this block's working set (84 KB) ----
  for (int c = tid; c < T_STEPS * 64; c += BLOCK) {  // 24 rows x 64 x 16B
    const int t = c >> 6;
    const int q = (c & 63) << 2;  // float index within row
    const long long g = (long long)t * D + d0 + q;
    async_load_b128(base + (unsigned int)((LDS_RJ + t * BLOCK + q) * 4), recog_J + g);
    async_load_b128(base + (unsigned int)((LDS_RH + t * BLOCK + q) * 4), recog_h + g);
  }
  for (int c = tid; c < S_ROWS * 64; c += BLOCK) {   // 12 rows x 64 x 16B
    const int s = c >> 6;
    const int q = (c & 63) << 2;
    const long long g = (long long)s * D + d0 + q;
    async_load_b128(base + (unsigned int)((LDS_AO + s * BLOCK + q) * 4), A + g);
    async_load_b128(base + (unsigned int)((LDS_LO + s * BLOCK + q) * 4), Lambda + g);
    async_load_b128(base + (unsigned int)((LDS_BO + s * BLOCK + q) * 4), b + g);
  }

  // ---- full RNG pipeline (threefry + erfinv) in the DMA shadow ----
  // Independent of staged data; keeps erfinv temporaries out of the
  // register-fat backward loop.
  float epsv[T_STEPS];
#pragma unroll
  for (int t = 0; t < S_ROWS; ++t) {  // counter pairs (t, t+12)
    unsigned int x0 = (unsigned int)t * (unsigned int)D + (unsigned int)d;
    unsigned int x1 = x0 + (unsigned int)S_ROWS * (unsigned int)D;  // N/2
    threefry2x32_00(x0, x1);
    epsv[t] = bits_to_normal(x0);
    epsv[t + S_ROWS] = bits_to_normal(x1);
  }

  wait_asynccnt0();
  __syncthreads();

#define RJ(t) smem[LDS_RJ + (t) * BLOCK + tid]
#define RH(t) smem[LDS_RH + (t) * BLOCK + tid]
#define AT(s) smem[LDS_AO + (s) * BLOCK + tid]
#define LT(s) smem[LDS_LO + (s) * BLOCK + tid]
#define BT(s) smem[LDS_BO + (s) * BLOCK + tid]

  // ---- forward information filter (state fully in VGPRs) ----
  float fJ[T_STEPS], fh[T_STEPS];
  const float tauv = __builtin_nontemporal_load(Tau + d);
  const float tau = tauv * tauv + EPS_F;
  const float locd = __builtin_nontemporal_load(loc + d);
  fJ[0] = tau + RJ(0);
  fh[0] = tau * locd + RH(0);
  float logZ = 0.0f;
  float plog = 0.0f;  // sum of lam*b^2 - log(lam) over the 23 transitions
#pragma unroll
  for (int t = 0; t < T_STEPS - 1; ++t) {
    const int s = (t < S_ROWS) ? t : (t - S_ROWS);  // _tile pattern
    const float l0 = LT(s);
    const float lam = l0 * l0 + EPS_F;
    const float av = AT(s);
    const float bv = BT(s);
    const float jab = av * lam;
    const float jaa = av * jab;
    const float ha_t = -(jab * bv);
    const float hb_t = lam * bv;
    const float denom = fJ[t] + jaa;
    const float inv = 1.0f / denom;
    const float hc = fh[t] + ha_t;
    logZ += 0.5f * (LOG_2PI_F - logf(denom)) + hc * hc * (0.5f * inv);
    fJ[t + 1] = lam - jab * jab * inv + RJ(t + 1);
    fh[t + 1] = hb_t + jab * hc * inv + RH(t + 1);
    plog += lam * bv * bv - logf(lam);
  }
  const float fJT = fJ[T_STEPS - 1];
  const float fhT = fh[T_STEPS - 1];
  const float q_logZ =
      logZ + 0.5f * (LOG_2PI_F - logf(fJT)) + fhT * fhT * (0.5f / fJT);
  const float p_logZ =
      0.5f * (LOG_2PI_F * (float)T_STEPS + tau * locd * locd - logf(tau) + plog);

  // ---- fused backward smoother + posterior sampler ----
  float Ex = fhT / fJT;
  float Var = 1.0f / fJT;
  float zc = Ex + epsv[T_STEPS - 1] / sqrtf(fJT);
  float kl = 0.0f;
  {
    const float Exx = Var + Ex * Ex;
    const long long o = (long long)(T_STEPS - 1) * D + d;
    __builtin_nontemporal_store(zc, z_out + o);
    __builtin_nontemporal_store(Ex, Ex_out + o);
    __builtin_nontemporal_store(Exx, Exx_out + o);
    kl += -0.5f * RJ(T_STEPS - 1) * Exx + RH(T_STEPS - 1) * Ex;
  }
#pragma unroll
  for (int t = T_STEPS - 2; t >= 0; --t) {
    const int s = (t < S_ROWS) ? t : (t - S_ROWS);
    const float l0 = LT(s);
    const float lam = l0 * l0 + EPS_F;
    const float av = AT(s);
    const float jab = av * lam;
    const float jaa = av * jab;
    const float ha_t = -(jab * BT(s));
    const float denom = fJ[t] + jaa;
    const float inv = 1.0f / denom;
    const float g = jab * inv;
    const float hca = fh[t] + ha_t;
    const float Ex_t = (hca + jab * Ex) * inv;
    Var = inv + g * g * Var;
    zc = (hca + jab * zc) * inv + epsv[t] / sqrtf(denom);
    Ex = Ex_t;
    const float Exx = Var + Ex * Ex;
    const long long o = (long long)t * D + d;
    __builtin_nontemporal_store(zc, z_out + o);
    __builtin_nontemporal_store(Ex, Ex_out + o);
    __builtin_nontemporal_store(Exx, Exx_out + o);
    kl += -0.5f * RJ(t) * Exx + RH(t) * Ex;
  }
  kl += p_logZ - q_logZ;

  // ---- deterministic per-block KL reduction (LDS tree) ----
  __syncthreads();
  smem[tid] = kl;
  __syncthreads();
#pragma unroll
  for (int off = BLOCK / 2; off > 0; off >>= 1) {
    if (tid < off) smem[tid] += smem[tid + off];
    __syncthreads();
  }
  if (tid == 0) kl_partial[blockIdx.x] = smem[0];

#undef RJ
#undef RH
#undef AT
#undef LT
#undef BT
}

__global__ __launch_bounds__(BLOCK) void kl_reduce_kernel(
    const float* __restrict__ partial, int n, float* __restrict__ out) {
  __shared__ float s[BLOCK];
  float acc = 0.0f;
  for (int i = threadIdx.x; i < n; i += BLOCK) acc += partial[i];
  s[threadIdx.x] = acc;
  __syncthreads();
  for (int off = BLOCK / 2; off > 0; off >>= 1) {
    if (threadIdx.x < off) s[threadIdx.x] += s[threadIdx.x + off];
    __syncthreads();
  }
  if (threadIdx.x == 0) out[0] = s[0];
}

extern "C" void kernel_launch(void* const* d_in, const int* in_sizes, int n_in,
                              void* d_out, int out_size, void* d_ws,
                              size_t ws_size, hipStream_t stream) {
  const float* recog_J = (const float*)d_in[0];  // (T, D)
  const float* recog_h = (const float*)d_in[1];  // (T, D)
  const float* locp    = (const float*)d_in[2];  // (D,)
  const float* Taup    = (const float*)d_in[3];  // (D,)
  const float* Lambdap = (const float*)d_in[4];  // (S, D)
  const float* Ap      = (const float*)d_in[5];  // (S, D)
  const float* bp      = (const float*)d_in[6];  // (S, D)
  const int D = in_sizes[2];                     // loc is (D,)

  float* out  = (float*)d_out;
  float* z    = out;                         // (T, D)
  float* klp  = out + (size_t)T_STEPS * D;   // scalar
  float* Exp  = klp + 1;                     // (T, D)
  float* Exxp = Exp + (size_t)T_STEPS * D;   // (T, D)
  float* partial = (float*)d_ws;

  const int nblocks = D / BLOCK;  // D = 262144 -> 1024 blocks
  pgm_chain_kernel<<<nblocks, BLOCK, 0, stream>>>(
      recog_J, recog_h, locp, Taup, Lambdap, Ap, bp, z, Exp, Exxp, partial, D);
  kl_reduce_kernel<<<1, BLOCK, 0, stream>>>(partial, nblocks, klp);
}